// Attention_9242769621486
// MI455X (gfx1250) — compile-verified
//
#include <hip/hip_runtime.h>
#include <math.h>

// ---------------- problem constants (from reference) ----------------
#define B_    2
#define CIN   64
#define CH    32      // q/k/v channels
#define T_    4
#define HH    32
#define WW    32
#define L_    1024    // H*W
#define HEADS 8
#define CPG   4       // channels per head (the WMMA K dimension!)
#define EPS   1e-5f

typedef __attribute__((ext_vector_type(2))) float v2f;
typedef __attribute__((ext_vector_type(8))) float v8f;

// ---------------- q conv: 1x3x3, pad (1,1) spatial ----------------
__global__ void k_conv_q(const float* __restrict__ x, const float* __restrict__ w,
                         const float* __restrict__ bias, float* __restrict__ q) {
  int idx = blockIdx.x * blockDim.x + threadIdx.x;
  if (idx >= B_ * CH * T_ * L_) return;
  int hw = idx & 1023;
  int t  = (idx >> 10) & 3;
  int co = (idx >> 12) & 31;
  int b  = idx >> 17;
  int h = hw >> 5, wx = hw & 31;
  float acc = bias[co];
  const float* xb = x + (((size_t)b * CIN) * T_ + t) * L_;  // + ci*T_*L_
  const float* wp = w + (size_t)co * CIN * 9;
  for (int ci = 0; ci < CIN; ++ci) {
    const float* xc = xb + (size_t)ci * (T_ * L_);
    const float* wc = wp + ci * 9;
#pragma unroll
    for (int dy = 0; dy < 3; ++dy) {
      int iy = h + dy - 1;
      if (iy < 0 || iy >= HH) continue;
#pragma unroll
      for (int dx = 0; dx < 3; ++dx) {
        int ix = wx + dx - 1;
        if (ix < 0 || ix >= WW) continue;
        acc = fmaf(xc[iy * WW + ix], wc[dy * 3 + dx], acc);
      }
    }
  }
  q[idx] = acc;
}

// ---------------- k & v convs: 1x1x1, fused ----------------
__global__ void k_conv_kv(const float* __restrict__ x,
                          const float* __restrict__ kw, const float* __restrict__ kb,
                          const float* __restrict__ vw, const float* __restrict__ vb,
                          float* __restrict__ k, float* __restrict__ v) {
  int idx = blockIdx.x * blockDim.x + threadIdx.x;
  if (idx >= B_ * CH * T_ * L_) return;
  int hw = idx & 1023;
  int t  = (idx >> 10) & 3;
  int co = (idx >> 12) & 31;
  int b  = idx >> 17;
  float ak = kb[co], av = vb[co];
  const float* xb = x + (((size_t)b * CIN) * T_ + t) * L_ + hw;
  const float* kwp = kw + co * CIN;
  const float* vwp = vw + co * CIN;
  for (int ci = 0; ci < CIN; ++ci) {
    float xv = xb[(size_t)ci * (T_ * L_)];
    ak = fmaf(xv, kwp[ci], ak);
    av = fmaf(xv, vwp[ci], av);
  }
  k[idx] = ak;
  v[idx] = av;
}

// ---------------- bilinear (align_corners) 64x64 -> 32x32 ----------------
__global__ void k_mask(const int* __restrict__ sm, float* __restrict__ m) {
  int idx = blockIdx.x * blockDim.x + threadIdx.x;
  if (idx >= B_ * L_) return;
  int xo = idx & 31, yo = (idx >> 5) & 31, b = idx >> 10;
  const float sc = 63.0f / 31.0f;
  float ys = yo * sc, xs = xo * sc;
  int y0 = (int)floorf(ys); int y1 = min(y0 + 1, 63); float fy = ys - (float)y0;
  int x0 = (int)floorf(xs); int x1 = min(x0 + 1, 63); float fx = xs - (float)x0;
  const int* sb = sm + (size_t)b * 64 * 64;
  float a00 = (float)sb[y0 * 64 + x0], a01 = (float)sb[y0 * 64 + x1];
  float a10 = (float)sb[y1 * 64 + x0], a11 = (float)sb[y1 * 64 + x1];
  float top = a00 * (1.0f - fx) + a01 * fx;
  float bot = a10 * (1.0f - fx) + a11 * fx;
  m[idx] = top * (1.0f - fy) + bot * fy;
}

// ---------------- fused attention per (b, head, t) ----------------
// scores[l,m] = sum_c q[c,l]*k[c,m]  via V_WMMA_F32_16X16X4_F32 (K=4 exact),
// masked exp (no max pass needed; |s| is O(5)), N=4 attn@V via VALU FMA,
// per-row reduction via wave32 shfl_xor within 16-lane halves.
// m-loop unrolled x4: 4 independent WMMA->exp->FMA chains hide WMMA result
// latency and LDS load latency behind each other.
__global__ __launch_bounds__(256) void k_attn(const float* __restrict__ q,
                                              const float* __restrict__ k,
                                              const float* __restrict__ v,
                                              const float* __restrict__ m,
                                              float* __restrict__ o) {
  __shared__ float sQ[CPG][L_];
  __shared__ float sK[CPG][L_];
  __shared__ float sV[CPG][L_];
  __shared__ float sM[L_];

  int bid = blockIdx.x;          // B*HEADS*T_*8 blocks
  int seg = bid & 7;
  int t   = (bid >> 3) & 3;
  int g   = (bid >> 5) & 7;
  int b   = bid >> 8;
  int tid = threadIdx.x;

  size_t base = (((size_t)b * CH + g * CPG) * T_ + t) * L_;  // + c*T_*L_ + l
  for (int i = tid; i < CPG * L_; i += 256) {
    int c = i >> 10, l = i & 1023;
    size_t off = base + (size_t)c * (T_ * L_) + l;
    sQ[c][l] = q[off];
    sK[c][l] = k[off];
    sV[c][l] = v[off];
  }
  for (int l = tid; l < L_; l += 256) sM[l] = m[(size_t)b * L_ + l];
  __syncthreads();

  int lane = tid & 31;
  int wave = tid >> 5;
  int hf   = lane >> 4;     // which half-wave (selects K-pair for A/B operands)
  int ln   = lane & 15;

  int lt = seg * 8 + wave;  // one 16-row l-tile per wave
  int l0 = lt << 4;

  // A operand (16x4 f32): lanes 0-15 hold K=0,1; lanes 16-31 hold K=2,3; M = ln
  v2f A;
  A.x = sQ[hf * 2 + 0][l0 + ln];
  A.y = sQ[hf * 2 + 1][l0 + ln];

  float rs[8], ac0[8], ac1[8], ac2[8], ac3[8];
#pragma unroll
  for (int r = 0; r < 8; ++r) { rs[r] = 0.f; ac0[r] = 0.f; ac1[r] = 0.f; ac2[r] = 0.f; ac3[r] = 0.f; }

#pragma unroll 4
  for (int mt = 0; mt < 64; ++mt) {
    int mc = (mt << 4) + ln;
    // B operand (4x16 f32): mirrors A layout; N = ln
    v2f Bv;
    Bv.x = sK[hf * 2 + 0][mc];
    Bv.y = sK[hf * 2 + 1][mc];
    v8f cz = {};
    v8f d = __builtin_amdgcn_wmma_f32_16x16x4_f32(
        /*neg_a=*/false, A, /*neg_b=*/false, Bv,
        /*c_mod=*/(short)0, cz, /*reuse_a=*/false, /*reuse_b=*/false);

    float mv = sM[mc];
    bool live = (mv != 0.0f);
    float v0 = sV[0][mc], v1 = sV[1][mc], v2 = sV[2][mc], v3 = sV[3][mc];
#pragma unroll
    for (int r = 0; r < 8; ++r) {
      float p = live ? __expf(d[r]) : 0.0f;
      rs[r] += p;
      ac0[r] = fmaf(p, v0, ac0[r]);
      ac1[r] = fmaf(p, v1, ac1[r]);
      ac2[r] = fmaf(p, v2, ac2[r]);
      ac3[r] = fmaf(p, v3, ac3[r]);
    }
  }

  // reduce over the 16 lanes holding each row (C/D layout: VGPR r, lanes0-15=M=r, lanes16-31=M=r+8)
#pragma unroll
  for (int off = 1; off < 16; off <<= 1) {
#pragma unroll
    for (int r = 0; r < 8; ++r) {
      rs[r]  += __shfl_xor(rs[r],  off, 32);
      ac0[r] += __shfl_xor(ac0[r], off, 32);
      ac1[r] += __shfl_xor(ac1[r], off, 32);
      ac2[r] += __shfl_xor(ac2[r], off, 32);
      ac3[r] += __shfl_xor(ac3[r], off, 32);
    }
  }

#pragma unroll
  for (int r = 0; r < 8; ++r) {
    if (ln == r) {
      int lrow = l0 + r + hf * 8;
      float inv = rs[r] > 0.0f ? 1.0f / rs[r] : 0.0f;
      size_t ob = base + (size_t)lrow;
      o[ob]                = ac0[r] * inv;
      o[ob + 1 * T_ * L_]  = ac1[r] * inv;
      o[ob + 2 * T_ * L_]  = ac2[r] * inv;
      o[ob + 3 * T_ * L_]  = ac3[r] * inv;
    }
  }
}

// ---------------- group-norm statistics (groups = 4) ----------------
__global__ void k_gn_stats(const float* __restrict__ src, float* __restrict__ stats,
                           int C, int cpg) {
  __shared__ float s1[256], s2[256];
  int bg = blockIdx.x;
  int b = bg >> 2, grp = bg & 3;
  size_t base = ((size_t)b * C + (size_t)grp * cpg) * (T_ * L_);
  int n = cpg * T_ * L_;
  float a = 0.f, a2 = 0.f;
  for (int i = threadIdx.x; i < n; i += 256) {
    float vv = src[base + i];
    a += vv; a2 = fmaf(vv, vv, a2);
  }
  s1[threadIdx.x] = a; s2[threadIdx.x] = a2;
  __syncthreads();
  for (int s = 128; s > 0; s >>= 1) {
    if (threadIdx.x < (unsigned)s) {
      s1[threadIdx.x] += s1[threadIdx.x + s];
      s2[threadIdx.x] += s2[threadIdx.x + s];
    }
    __syncthreads();
  }
  if (threadIdx.x == 0) {
    float mean = s1[0] / (float)n;
    float var  = s2[0] / (float)n - mean * mean;
    stats[bg * 2]     = mean;
    stats[bg * 2 + 1] = rsqrtf(var + EPS);
  }
}

// ---------------- gn apply + relu ----------------
__global__ void k_gn_relu(const float* __restrict__ src, const float* __restrict__ stats,
                          const float* __restrict__ w, const float* __restrict__ bias,
                          float* __restrict__ dst, int C, int cpg) {
  int idx = blockIdx.x * blockDim.x + threadIdx.x;
  if (idx >= B_ * C * T_ * L_) return;
  int ch = (idx / (T_ * L_)) % C;
  int b  = idx / (C * T_ * L_);
  int grp = ch / cpg;
  float mean = stats[(b * 4 + grp) * 2];
  float rstd = stats[(b * 4 + grp) * 2 + 1];
  float h = fmaf((src[idx] - mean) * rstd, w[ch], bias[ch]);
  dst[idx] = fmaxf(h, 0.0f);
}

// ---------------- agg conv 1x1x1: 32 -> 64 ----------------
__global__ void k_conv1x1(const float* __restrict__ src, const float* __restrict__ w,
                          float* __restrict__ dst) {
  int idx = blockIdx.x * blockDim.x + threadIdx.x;
  if (idx >= B_ * 64 * T_ * L_) return;
  int tl = idx % (T_ * L_);
  int co = (idx / (T_ * L_)) % 64;
  int b  = idx / (64 * T_ * L_);
  const float* sp = src + ((size_t)b * CH) * (T_ * L_) + tl;
  const float* wp = w + co * CH;
  float acc = 0.f;
#pragma unroll 8
  for (int ci = 0; ci < CH; ++ci) acc = fmaf(sp[(size_t)ci * (T_ * L_)], wp[ci], acc);
  dst[idx] = acc;
}

// ---------------- gn2 apply + relu + residual add ----------------
__global__ void k_gn_relu_add(const float* __restrict__ h2, const float* __restrict__ stats,
                              const float* __restrict__ w, const float* __restrict__ bias,
                              const float* __restrict__ x, float* __restrict__ dst) {
  int idx = blockIdx.x * blockDim.x + threadIdx.x;
  if (idx >= B_ * 64 * T_ * L_) return;
  int ch = (idx / (T_ * L_)) % 64;
  int b  = idx / (64 * T_ * L_);
  int grp = ch >> 4;  // cpg = 16
  float mean = stats[(b * 4 + grp) * 2];
  float rstd = stats[(b * 4 + grp) * 2 + 1];
  float h = fmaf((h2[idx] - mean) * rstd, w[ch], bias[ch]);
  dst[idx] = fmaxf(h, 0.0f) + x[idx];
}

// ---------------- final gn apply (in place on d_out) ----------------
__global__ void k_gn_apply(float* __restrict__ buf, const float* __restrict__ stats,
                           const float* __restrict__ w, const float* __restrict__ bias) {
  int idx = blockIdx.x * blockDim.x + threadIdx.x;
  if (idx >= B_ * 64 * T_ * L_) return;
  int ch = (idx / (T_ * L_)) % 64;
  int b  = idx / (64 * T_ * L_);
  int grp = ch >> 4;
  float mean = stats[(b * 4 + grp) * 2];
  float rstd = stats[(b * 4 + grp) * 2 + 1];
  buf[idx] = fmaf((buf[idx] - mean) * rstd, w[ch], bias[ch]);
}

extern "C" void kernel_launch(void* const* d_in, const int* in_sizes, int n_in,
                              void* d_out, int out_size, void* d_ws, size_t ws_size,
                              hipStream_t stream) {
  const float* x    = (const float*)d_in[0];
  const int*   sm   = (const int*)d_in[1];
  const float* qw   = (const float*)d_in[2];
  const float* qb   = (const float*)d_in[3];
  const float* kw   = (const float*)d_in[4];
  const float* kb   = (const float*)d_in[5];
  const float* vw   = (const float*)d_in[6];
  const float* vb   = (const float*)d_in[7];
  const float* gn1w = (const float*)d_in[8];
  const float* gn1b = (const float*)d_in[9];
  const float* aw   = (const float*)d_in[10];
  const float* gn2w = (const float*)d_in[11];
  const float* gn2b = (const float*)d_in[12];
  const float* ognw = (const float*)d_in[13];
  const float* ognb = (const float*)d_in[14];
  float* out = (float*)d_out;

  // workspace carve-up (~7.3 MB of fp32)
  float* ws = (float*)d_ws;
  const size_t NQ = (size_t)B_ * CH * T_ * L_;  // 262144
  const size_t NO = (size_t)B_ * 64 * T_ * L_;  // 524288
  float* qbuf  = ws; ws += NQ;
  float* kbuf  = ws; ws += NQ;
  float* vbuf  = ws; ws += NQ;
  float* obuf  = ws; ws += NQ;
  float* mbuf  = ws; ws += (size_t)B_ * L_;
  float* hbuf  = ws; ws += NQ;
  float* h2buf = ws; ws += NO;
  float* st1   = ws; ws += 16;
  float* st2   = ws; ws += 16;
  float* st3   = ws; ws += 16;

  k_conv_q<<<(int)(NQ / 256), 256, 0, stream>>>(x, qw, qb, qbuf);
  k_conv_kv<<<(int)(NQ / 256), 256, 0, stream>>>(x, kw, kb, vw, vb, kbuf, vbuf);
  k_mask<<<(B_ * L_) / 256, 256, 0, stream>>>(sm, mbuf);

  // 512 workgroups: (b, head, t) x 8 l-segments, 8 waves each, 1 l-tile/wave
  k_attn<<<B_ * HEADS * T_ * 8, 256, 0, stream>>>(qbuf, kbuf, vbuf, mbuf, obuf);

  k_gn_stats<<<8, 256, 0, stream>>>(obuf, st1, CH, 8);
  k_gn_relu<<<(int)(NQ / 256), 256, 0, stream>>>(obuf, st1, gn1w, gn1b, hbuf, CH, 8);
  k_conv1x1<<<(int)(NO / 256), 256, 0, stream>>>(hbuf, aw, h2buf);
  k_gn_stats<<<8, 256, 0, stream>>>(h2buf, st2, 64, 16);
  k_gn_relu_add<<<(int)(NO / 256), 256, 0, stream>>>(h2buf, st2, gn2w, gn2b, x, out);
  k_gn_stats<<<8, 256, 0, stream>>>(out, st3, 64, 16);
  k_gn_apply<<<(int)(NO / 256), 256, 0, stream>>>(out, st3, ognw, ognb);
}